// LSTM_67070209294915
// MI455X (gfx1250) — compile-verified
//
#include <hip/hip_runtime.h>

typedef __attribute__((ext_vector_type(2))) float v2f;
typedef __attribute__((ext_vector_type(8))) float v8f;

#define B_   512
#define T_   2048
#define E_   8
#define H_   2
#define OUT_ 128
#define FC_TILES_PER_WAVE 8   // row-tiles (of 16 rows) per wave in fc_wmma

__device__ __forceinline__ float sigmoidf(float x) {
    return 1.0f / (1.0f + __expf(-x));
}

// ---------------------------------------------------------------------------
// Kernel 1: embedding gather + input-gate precompute (fully parallel):
//   gx0[r, g] = b0[g] + sum_d W_ih0[g,d] * emb[x[r], d]     r in [0, B*T)
// Default (RT) stores: gx0 (32 MB) is re-read by the scan and fits in L2.
// ---------------------------------------------------------------------------
__global__ void embed_gates0(const int* __restrict__ x,
                             const float* __restrict__ emb,
                             const float* __restrict__ W_ih0,
                             const float* __restrict__ b0,
                             float* __restrict__ gx0, int n) {
    __shared__ float sW[64];
    __shared__ float sb[8];
    int tid = threadIdx.x;
    if (tid < 64) sW[tid] = W_ih0[tid];
    if (tid < 8)  sb[tid] = b0[tid];
    __syncthreads();

    int r = blockIdx.x * blockDim.x + tid;
    if (r >= n) return;

    int tok = x[r];
    const float* e = emb + (long)tok * E_;
    float ev[8];
#pragma unroll
    for (int d = 0; d < 8; ++d) ev[d] = e[d];

    float g[8];
#pragma unroll
    for (int j = 0; j < 8; ++j) {
        float acc = sb[j];
#pragma unroll
        for (int d = 0; d < 8; ++d) acc = fmaf(sW[j * 8 + d], ev[d], acc);
        g[j] = acc;
    }
    float4* o4 = (float4*)(gx0 + (long)r * 8);
    o4[0] = make_float4(g[0], g[1], g[2], g[3]);
    o4[1] = make_float4(g[4], g[5], g[6], g[7]);
}

// ---------------------------------------------------------------------------
// Kernel 2: the serial scan. One thread per batch row; weights in registers.
// Writes y1[b,t,0:2] and the final h_n / c_n straight into d_out's tail.
// ---------------------------------------------------------------------------
__global__ void lstm_scan(const float* __restrict__ gx0,
                          const float* __restrict__ W_hh0,
                          const float* __restrict__ W_ih1,
                          const float* __restrict__ W_hh1,
                          const float* __restrict__ b1,
                          float* __restrict__ y1,
                          float* __restrict__ hn_cn /* = d_out + B*T*OUT */) {
    int b = blockIdx.x * blockDim.x + threadIdx.x;
    if (b >= B_) return;

    float whh0[16], wih1[16], whh1[16], bb1[8];
#pragma unroll
    for (int i = 0; i < 16; ++i) { whh0[i] = W_hh0[i]; wih1[i] = W_ih1[i]; whh1[i] = W_hh1[i]; }
#pragma unroll
    for (int i = 0; i < 8; ++i) bb1[i] = b1[i];

    float h0[2] = {0.f, 0.f}, c0[2] = {0.f, 0.f};
    float h1[2] = {0.f, 0.f}, c1[2] = {0.f, 0.f};

    const float* gxp = gx0 + (long)b * T_ * 8;
    float*       yp  = y1  + (long)b * T_ * 2;

    for (int t = 0; t < T_; ++t) {
        // prefetch the gates stream ahead (global_prefetch_b8)
        int tp = (t + 16 < T_) ? (t + 16) : (T_ - 1);
        __builtin_prefetch(gxp + (long)tp * 8, 0, 0);

        float4 ga = ((const float4*)gxp)[t * 2 + 0];
        float4 gb = ((const float4*)gxp)[t * 2 + 1];
        float g0[8] = {ga.x, ga.y, ga.z, ga.w, gb.x, gb.y, gb.z, gb.w};
#pragma unroll
        for (int j = 0; j < 8; ++j)
            g0[j] = fmaf(whh0[j * 2 + 0], h0[0], fmaf(whh0[j * 2 + 1], h0[1], g0[j]));
#pragma unroll
        for (int k = 0; k < 2; ++k) {
            float ig = sigmoidf(g0[0 + k]);
            float fg = sigmoidf(g0[2 + k]);
            float gg = tanhf(g0[4 + k]);
            float og = sigmoidf(g0[6 + k]);
            c0[k] = fg * c0[k] + ig * gg;
            h0[k] = og * tanhf(c0[k]);
        }

        float g1[8];
#pragma unroll
        for (int j = 0; j < 8; ++j)
            g1[j] = bb1[j]
                  + wih1[j * 2 + 0] * h0[0] + wih1[j * 2 + 1] * h0[1]
                  + whh1[j * 2 + 0] * h1[0] + whh1[j * 2 + 1] * h1[1];
#pragma unroll
        for (int k = 0; k < 2; ++k) {
            float ig = sigmoidf(g1[0 + k]);
            float fg = sigmoidf(g1[2 + k]);
            float gg = tanhf(g1[4 + k]);
            float og = sigmoidf(g1[6 + k]);
            c1[k] = fg * c1[k] + ig * gg;
            h1[k] = og * tanhf(c1[k]);
        }
        ((float2*)yp)[t] = make_float2(h1[0], h1[1]);
    }

    // h_n: [2, B, 2] then c_n: [2, B, 2], flattened after out
    hn_cn[0 * B_ * 2 + b * 2 + 0] = h0[0];
    hn_cn[0 * B_ * 2 + b * 2 + 1] = h0[1];
    hn_cn[1 * B_ * 2 + b * 2 + 0] = h1[0];
    hn_cn[1 * B_ * 2 + b * 2 + 1] = h1[1];
    float* cn = hn_cn + 2 * B_ * 2;
    cn[0 * B_ * 2 + b * 2 + 0] = c0[0];
    cn[0 * B_ * 2 + b * 2 + 1] = c0[1];
    cn[1 * B_ * 2 + b * 2 + 0] = c1[0];
    cn[1 * B_ * 2 + b * 2 + 1] = c1[1];
}

// ---------------------------------------------------------------------------
// Kernel 3: out[r, n] = sum_k y1[r,k] * W_fc[n,k] + b_fc[n]
// via V_WMMA_F32_16X16X4_F32 (K=2 zero-padded to 4; exact f32 math).
// One wave -> FC_TILES_PER_WAVE 16x16 tiles (B-fragment + bias loaded once);
// 8 waves per block cover all 128 columns. Output written with non-temporal
// stores: the 512 MB stream is never re-read and must not evict the L2-resident
// workspace (gx0 + y1 = 40 MB << 192 MB L2).
// ---------------------------------------------------------------------------
__global__ void fc_wmma(const float* __restrict__ y1,
                        const float* __restrict__ W_fc,
                        const float* __restrict__ b_fc,
                        float* __restrict__ out) {
    int lane = threadIdx.x & 31;
    int wave = threadIdx.x >> 5;            // 0..7 : column tile
    int l15 = lane & 15;
    int n = wave * 16 + l15;                // output column for this lane
    long rowBase0 = (long)blockIdx.x * (16 * FC_TILES_PER_WAVE);

    // B: 4x16 f32, B[k][n] = W_fc[n][k]; rows K2,K3 are the zero pad.
    v2f bm;
    if (lane < 16) {
        float2 wv = ((const float2*)W_fc)[n];
        bm.x = wv.x; bm.y = wv.y;
    } else {
        bm.x = 0.f; bm.y = 0.f;
    }

    // C: bias broadcast (per-column value, identical across the 8 row-VGPRs)
    float bias = b_fc[n];
    v8f cbias;
#pragma unroll
    for (int j = 0; j < 8; ++j) cbias[j] = bias;

    int mOff = (lane >= 16) ? 8 : 0;

    for (int it = 0; it < FC_TILES_PER_WAVE; ++it) {
        long rowBase = rowBase0 + (long)it * 16;

        // A: 16x4 f32. VGPR0 = K0 (lanes0-15) | K2 (lanes16-31, zero pad);
        //              VGPR1 = K1            | K3 (zero pad).
        v2f a;
        if (lane < 16) {
            float2 yv = ((const float2*)y1)[rowBase + lane];
            a.x = yv.x; a.y = yv.y;
        } else {
            a.x = 0.f; a.y = 0.f;
        }

        v8f d = __builtin_amdgcn_wmma_f32_16x16x4_f32(
            /*neg_a=*/false, a, /*neg_b=*/false, bm,
            /*c_mod=*/(short)0, cbias, /*reuse_a=*/false, /*reuse_b=*/false);

        // D layout: VGPR j holds row M=j (lanes0-15) / M=j+8 (lanes16-31).
        float* op = out + (rowBase + mOff) * OUT_ + n;
#pragma unroll
        for (int j = 0; j < 8; ++j)
            __builtin_nontemporal_store(d[j], op + (long)j * OUT_);
    }
}

// ---------------------------------------------------------------------------
extern "C" void kernel_launch(void* const* d_in, const int* in_sizes, int n_in,
                              void* d_out, int out_size, void* d_ws, size_t ws_size,
                              hipStream_t stream) {
    const int*   x     = (const int*)d_in[0];
    const float* emb   = (const float*)d_in[1];
    const float* W_ih0 = (const float*)d_in[2];
    const float* W_hh0 = (const float*)d_in[3];
    const float* b0    = (const float*)d_in[4];
    const float* W_ih1 = (const float*)d_in[5];
    const float* W_hh1 = (const float*)d_in[6];
    const float* b1    = (const float*)d_in[7];
    const float* W_fc  = (const float*)d_in[8];
    const float* b_fc  = (const float*)d_in[9];
    float* out = (float*)d_out;

    float* gx0 = (float*)d_ws;                     // [B*T, 8]  = 32 MB
    float* y1  = gx0 + (size_t)B_ * T_ * 8;        // [B*T, 2]  =  8 MB

    const int n = B_ * T_;                         // 1,048,576 rows

    embed_gates0<<<n / 256, 256, 0, stream>>>(x, emb, W_ih0, b0, gx0, n);

    lstm_scan<<<(B_ + 255) / 256, 256, 0, stream>>>(
        gx0, W_hh0, W_ih1, W_hh1, b1, y1, out + (size_t)n * OUT_);

    fc_wmma<<<n / (16 * FC_TILES_PER_WAVE), 256, 0, stream>>>(y1, W_fc, b_fc, out);
}